// GAT_17506286698971
// MI455X (gfx1250) — compile-verified
//
#include <hip/hip_runtime.h>

#define NN      50000      // nodes
#define NE      800000     // edges (without self loops)
#define ET      (NE + NN)  // edges + self loops
#define HEADS   4
#define C_HID   32
#define C_EMB   128
#define OUT_DIM 2
#define F1      (HEADS * C_HID)   // 128
#define F2      (HEADS * C_EMB)   // 512
#define NEG_SLOPE 0.2f

typedef float v2f __attribute__((ext_vector_type(2)));
typedef float v8f __attribute__((ext_vector_type(8)));

// ---------- order-preserving float<->uint for atomic segment max ----------
__device__ __forceinline__ unsigned enc_ord(float f) {
    unsigned u = __float_as_uint(f);
    return (u & 0x80000000u) ? ~u : (u | 0x80000000u);
}
__device__ __forceinline__ float dec_ord(unsigned v) {
    return __uint_as_float((v & 0x80000000u) ? (v ^ 0x80000000u) : ~v);
}
#define ENC_NEG_INF 0x007FFFFFu   // enc_ord(-inf)

// ---------- generic u32 fill (also zeroes f32 buffers) ----------
__global__ void fill_u32(unsigned* __restrict__ p, unsigned v, long long count) {
    long long i = (long long)blockIdx.x * blockDim.x + threadIdx.x;
    long long stride = (long long)gridDim.x * blockDim.x;
    for (; i < count; i += stride) p[i] = v;
}

// ---------- WMMA f32 GEMM: C[M,N] = A[M,K] @ B[K,N], row-major ----------
// One wave computes a 16(M) x 64(N) strip: A fragment loaded once per k-step,
// reused by 4 chained WMMAs. All B fragments are loaded into distinct regs
// BEFORE the WMMA chain so the loads form one clause and the waits stagger.
// Requires: K % 4 == 0, M % 16 == 0, N % 64 == 0.
__global__ void gemm_f32_wmma(const float* __restrict__ A,
                              const float* __restrict__ B,
                              float* __restrict__ C,
                              int M, int N, int K) {
    const int lane = threadIdx.x & 31;
    const int wv   = threadIdx.x >> 5;
    const int tm   = (blockIdx.y * 8 + wv) * 16;
    const int tn   = blockIdx.x * 64;
    if (tm >= M || tn >= N) return;   // wave-uniform: EXEC all-ones at WMMAs

    const int r    = lane & 15;   // M row (A) / N col (B,C)
    const int half = lane >> 4;   // selects K pair {0,1} vs {2,3}

    const float* Ap = A + (size_t)(tm + r) * K + half * 2;
    const float* Bp = B + (size_t)(half * 2) * N + tn + r;

    v8f acc0 = {0.f,0.f,0.f,0.f,0.f,0.f,0.f,0.f};
    v8f acc1 = acc0, acc2 = acc0, acc3 = acc0;

    for (int k = 0; k < K; k += 4) {
        const float* b0 = Bp + (size_t)k * N;     // row kb
        const float* b1 = b0 + N;                 // row kb+1
        // ---- issue all loads first (separate temporaries -> one clause) ----
        v2f a = *(const v2f*)(Ap + k);            // A[m][kb], A[m][kb+1]
        v2f bf0, bf1, bf2, bf3;
        bf0.x = b0[0];  bf0.y = b1[0];
        bf1.x = b0[16]; bf1.y = b1[16];
        bf2.x = b0[32]; bf2.y = b1[32];
        bf3.x = b0[48]; bf3.y = b1[48];
        // ---- 4 WMMAs reusing the same A fragment ----
        acc0 = __builtin_amdgcn_wmma_f32_16x16x4_f32(false, a, false, bf0, (short)0, acc0, false, false);
        acc1 = __builtin_amdgcn_wmma_f32_16x16x4_f32(false, a, false, bf1, (short)0, acc1, false, false);
        acc2 = __builtin_amdgcn_wmma_f32_16x16x4_f32(false, a, false, bf2, (short)0, acc2, false, false);
        acc3 = __builtin_amdgcn_wmma_f32_16x16x4_f32(false, a, false, bf3, (short)0, acc3, false, false);
    }
    // C/D layout: VGPR i -> row tm + i + half*8, col tn + 16*t + (lane&15)
    float* Cp = C + (size_t)(tm + half * 8) * N + tn + r;
#pragma unroll
    for (int i = 0; i < 8; ++i) {
        float* row = Cp + (size_t)i * N;
        row[0]  = acc0[i];
        row[16] = acc1[i];
        row[32] = acc2[i];
        row[48] = acc3[i];
    }
}

// ---------- edge pass 1: attention logits + segment max ----------
// one wave per edge; 8 lanes per head (works for F=128 and F=512)
template <int F>
__global__ void edge_pass1(const float* __restrict__ xl,
                           const float* __restrict__ xr,
                           const int*  __restrict__ src,
                           const int*  __restrict__ dst,
                           const float* __restrict__ att,
                           float* __restrict__ eb,
                           unsigned* __restrict__ emax) {
    constexpr int VEC = F / 32;
    const int e = (int)((blockIdx.x * blockDim.x + threadIdx.x) >> 5);
    if (e >= ET) return;
    const int lane = threadIdx.x & 31;
    int s, d;
    if (e < NE) { s = src[e]; d = dst[e]; } else { s = d = e - NE; }

    const float4* xlp = (const float4*)(xl + (size_t)s * F + lane * VEC);
    const float4* xrp = (const float4*)(xr + (size_t)d * F + lane * VEC);
    const float4* ap  = (const float4*)(att + lane * VEC);

    float sum = 0.f;
#pragma unroll
    for (int v = 0; v < VEC / 4; ++v) {
        float4 a = xlp[v], b = xrp[v], w = ap[v];
        float m;
        m = a.x + b.x; sum += (m > 0.f ? m : NEG_SLOPE * m) * w.x;
        m = a.y + b.y; sum += (m > 0.f ? m : NEG_SLOPE * m) * w.y;
        m = a.z + b.z; sum += (m > 0.f ? m : NEG_SLOPE * m) * w.z;
        m = a.w + b.w; sum += (m > 0.f ? m : NEG_SLOPE * m) * w.w;
    }
    // reduce within each 8-lane head group
    sum += __shfl_xor(sum, 1, 32);
    sum += __shfl_xor(sum, 2, 32);
    sum += __shfl_xor(sum, 4, 32);
    if ((lane & 7) == 0) {
        const int h = lane >> 3;
        eb[(size_t)e * HEADS + h] = sum;
        atomicMax(&emax[(size_t)d * HEADS + h], enc_ord(sum));
    }
}

// ---------- edge pass 2: exp, denominator, weighted aggregate ----------
template <int F>
__global__ void edge_pass2(const float* __restrict__ xl,
                           const int*  __restrict__ src,
                           const int*  __restrict__ dst,
                           const float* __restrict__ eb,
                           const unsigned* __restrict__ emax,
                           float* __restrict__ denom,
                           float* __restrict__ agg) {
    constexpr int VEC = F / 32;
    const int e = (int)((blockIdx.x * blockDim.x + threadIdx.x) >> 5);
    if (e >= ET) return;
    const int lane = threadIdx.x & 31;
    int s, d;
    if (e < NE) { s = src[e]; d = dst[e]; } else { s = d = e - NE; }

    const int h  = lane >> 3;
    const float ev = eb[(size_t)e * HEADS + h];
    const float mx = dec_ord(emax[(size_t)d * HEADS + h]);
    const float ex = __expf(ev - mx);
    if ((lane & 7) == 0) atomicAdd(&denom[(size_t)d * HEADS + h], ex);

    const float4* xlp = (const float4*)(xl + (size_t)s * F + lane * VEC);
    float* ag = agg + (size_t)d * F + lane * VEC;
#pragma unroll
    for (int v = 0; v < VEC / 4; ++v) {
        float4 a = xlp[v];
        atomicAdd(ag + 4 * v + 0, ex * a.x);
        atomicAdd(ag + 4 * v + 1, ex * a.y);
        atomicAdd(ag + 4 * v + 2, ex * a.z);
        atomicAdd(ag + 4 * v + 3, ex * a.w);
    }
}

// ---------- layer-1 finalize: normalize, +bias, relu (in place) ----------
__global__ void finalize_concat_relu(float* __restrict__ agg,
                                     const float* __restrict__ denom,
                                     const float* __restrict__ bias) {
    int idx = blockIdx.x * blockDim.x + threadIdx.x;
    if (idx >= NN * F1) return;
    int n = idx / F1, f = idx - n * F1, h = f / C_HID;
    float v = agg[idx] / denom[n * HEADS + h] + bias[f];
    agg[idx] = v > 0.f ? v : 0.f;
}

// ---------- layer-2 finalize: normalize, mean over heads, +bias ----------
__global__ void finalize_mean(const float* __restrict__ agg,   // [N,H,C_EMB]
                              const float* __restrict__ denom, // [N,H]
                              const float* __restrict__ bias,  // [C_EMB]
                              float* __restrict__ hout) {      // [N,C_EMB]
    int idx = blockIdx.x * blockDim.x + threadIdx.x;
    if (idx >= NN * C_EMB) return;
    int n = idx / C_EMB, c = idx - n * C_EMB;
    float acc = 0.f;
#pragma unroll
    for (int h = 0; h < HEADS; ++h)
        acc += agg[((size_t)n * HEADS + h) * C_EMB + c] / denom[n * HEADS + h];
    hout[idx] = acc * (1.0f / HEADS) + bias[c];
}

// ---------- decoder: out = h @ Wo + bo  (OUT_DIM=2, trivial) ----------
__global__ void decoder_k(const float* __restrict__ h,
                          const float* __restrict__ Wo,
                          const float* __restrict__ bo,
                          float* __restrict__ out) {
    int n = blockIdx.x * blockDim.x + threadIdx.x;
    if (n >= NN) return;
    const float* hp = h + (size_t)n * C_EMB;
    float a0 = 0.f, a1 = 0.f;
#pragma unroll 4
    for (int k = 0; k < C_EMB; ++k) {
        float v = hp[k];
        a0 += v * Wo[2 * k + 0];
        a1 += v * Wo[2 * k + 1];
    }
    out[2 * n + 0] = a0 + bo[0];
    out[2 * n + 1] = a1 + bo[1];
}

extern "C" void kernel_launch(void* const* d_in, const int* in_sizes, int n_in,
                              void* d_out, int out_size, void* d_ws, size_t ws_size,
                              hipStream_t stream) {
    (void)in_sizes; (void)n_in; (void)out_size; (void)ws_size;

    const float* x    = (const float*)d_in[0];
    const int*   ei   = (const int*)d_in[1];       // edge_index [2,NE]
    const float* Wl1  = (const float*)d_in[2];
    const float* Wr1  = (const float*)d_in[3];
    const float* att1 = (const float*)d_in[4];
    const float* b1   = (const float*)d_in[5];
    const float* Wl2  = (const float*)d_in[6];
    const float* Wr2  = (const float*)d_in[7];
    const float* att2 = (const float*)d_in[8];
    const float* b2   = (const float*)d_in[9];
    const float* Wo   = (const float*)d_in[10];
    const float* bo   = (const float*)d_in[11];

    const int* src = ei;
    const int* dst = ei + NE;

    // ---- workspace layout (floats) ----
    float* ws = (float*)d_ws;
    size_t o = 0;
    float* xl1  = ws + o; o += (size_t)NN * F1;
    float* xr1  = ws + o; o += (size_t)NN * F1;
    float* agg1 = ws + o; o += (size_t)NN * F1;   // becomes h1 after finalize
    float* xl2  = ws + o; o += (size_t)NN * F2;
    float* xr2  = ws + o; o += (size_t)NN * F2;
    float* agg2 = ws + o; o += (size_t)NN * F2;
    float* eb   = ws + o; o += (size_t)ET * HEADS;
    unsigned* emaxu = (unsigned*)(ws + o); o += (size_t)NN * HEADS;
    float* denom = ws + o; o += (size_t)NN * HEADS;

    float* out_final = (float*)d_out;                         // [NN, 2]
    float* h_final   = (float*)d_out + (size_t)NN * OUT_DIM;  // [NN, 128]

    const dim3 blk(256);
    const dim3 eg((ET + 7) / 8);     // 8 edges (one wave each) per 256-thread block
    const dim3 fg(1024);

    const int mtiles8 = (NN / 16 + 7) / 8;   // 8 M-tiles (waves) per block

    // ===================== layer 1 =====================
    { dim3 g(F1 / 64, mtiles8);
      gemm_f32_wmma<<<g, blk, 0, stream>>>(x, Wl1, xl1, NN, F1, 128);
      gemm_f32_wmma<<<g, blk, 0, stream>>>(x, Wr1, xr1, NN, F1, 128); }

    fill_u32<<<fg, blk, 0, stream>>>(emaxu, ENC_NEG_INF, (long long)NN * HEADS);
    fill_u32<<<fg, blk, 0, stream>>>((unsigned*)denom, 0u, (long long)NN * HEADS);
    fill_u32<<<fg, blk, 0, stream>>>((unsigned*)agg1, 0u, (long long)NN * F1);

    edge_pass1<F1><<<eg, blk, 0, stream>>>(xl1, xr1, src, dst, att1, eb, emaxu);
    edge_pass2<F1><<<eg, blk, 0, stream>>>(xl1, src, dst, eb, emaxu, denom, agg1);
    finalize_concat_relu<<<(NN * F1 + 255) / 256, blk, 0, stream>>>(agg1, denom, b1);

    // ===================== layer 2 =====================
    { dim3 g(F2 / 64, mtiles8);
      gemm_f32_wmma<<<g, blk, 0, stream>>>(agg1, Wl2, xl2, NN, F2, F1);
      gemm_f32_wmma<<<g, blk, 0, stream>>>(agg1, Wr2, xr2, NN, F2, F1); }

    fill_u32<<<fg, blk, 0, stream>>>(emaxu, ENC_NEG_INF, (long long)NN * HEADS);
    fill_u32<<<fg, blk, 0, stream>>>((unsigned*)denom, 0u, (long long)NN * HEADS);
    fill_u32<<<fg, blk, 0, stream>>>((unsigned*)agg2, 0u, (long long)NN * F2);

    edge_pass1<F2><<<eg, blk, 0, stream>>>(xl2, xr2, src, dst, att2, eb, emaxu);
    edge_pass2<F2><<<eg, blk, 0, stream>>>(xl2, src, dst, eb, emaxu, denom, agg2);
    finalize_mean<<<(NN * C_EMB + 255) / 256, blk, 0, stream>>>(agg2, denom, b2, h_final);

    // ===================== decoder =====================
    decoder_k<<<(NN + 255) / 256, blk, 0, stream>>>(h_final, Wo, bo, out_final);
}